// LinearSelfAttentionBlock_24223615550348
// MI455X (gfx1250) — compile-verified
//
#include <hip/hip_runtime.h>
#include <hip/hip_bf16.h>
#include <math.h>

// ---------------------------------------------------------------------------
// LinearSelfAttentionBlock forward for MI455X (gfx1250, wave32, WMMA).
// GEMMs use V_WMMA_F32_16X16X4_F32 (fp32: workload is HBM-bandwidth bound at
// 23.3 TB/s, so full-precision matrix ops cost nothing extra). Tile staging
// uses GLOBAL_LOAD_ASYNC_TO_LDS_B128 (ASYNCcnt) where the copy is reformat-
// free, with a portable fallback when the builtin is unavailable.
// ---------------------------------------------------------------------------

typedef float v2f __attribute__((ext_vector_type(2)));
typedef float v8f __attribute__((ext_vector_type(8)));
typedef int   v4i_t __attribute__((vector_size(16)));   // matches builtin param

#define ACT_NONE  0
#define ACT_TANH1 1
#define ACT_GELU  2

#if __has_builtin(__builtin_amdgcn_global_load_async_to_lds_b128)
#define HAVE_ASYNC_LDS 1
#else
#define HAVE_ASYNC_LDS 0
#endif

// copy 16 bytes global -> LDS (per active lane)
__device__ __forceinline__ void async_copy16(const float* __restrict__ g, float* l)
{
#if HAVE_ASYNC_LDS
    __builtin_amdgcn_global_load_async_to_lds_b128(
        (__attribute__((address_space(1))) v4i_t*)g,
        (__attribute__((address_space(3))) v4i_t*)l, 0, 0);
#else
    const float4 v = *(const float4*)g;
    *(float4*)l = v;
#endif
}

// wait for outstanding async copies, then block barrier
__device__ __forceinline__ void async_join()
{
#if HAVE_ASYNC_LDS
#if __has_builtin(__builtin_amdgcn_s_wait_asynccnt)
    __builtin_amdgcn_s_wait_asynccnt(0);
#else
    asm volatile("s_wait_asynccnt 0" ::: "memory");
#endif
#endif
    __syncthreads();
}

// ---------------------------------------------------------------------------
// Generic C = act(A[M,K] @ W[K,N] + bias[N]) (+ r1) (+ r2)
// Block tile 64x64, BK=32. 256 threads = 8 waves; wave (wr,wc) computes a
// 16x32 sub-tile as two 16x16 WMMA accumulators. M,N,K multiples of 64/32.
// fp32 WMMA fragment layouts (CDNA5 ISA 7.12.2):
//   A 16x4:  lane l holds A[m = l&15][k = vgpr + 2*(l>>4)]   (2 VGPRs)
//   B 4x16:  lane l holds B[k = vgpr + 2*(l>>4)][n = l&15]   (2 VGPRs)
//   C 16x16: lane l, vgpr v holds C[m = v + 8*(l>>4)][n = l&15]
// A tile: row-major + pad 4 (16B-aligned rows for async b128, 36 stride has
// no 64-bank column conflicts). W tile: k-paired layout Wp[k/2][n][k&1] so a
// B fragment is one aligned ds_load_b64 (no shuffle movs in the hot loop).
// ---------------------------------------------------------------------------
__global__ __launch_bounds__(256)
void gemm_f32_wmma_kernel(const float* __restrict__ A,
                          const float* __restrict__ W,
                          const float* __restrict__ bias,
                          const float* __restrict__ r1,
                          const float* __restrict__ r2,
                          float* __restrict__ C,
                          int M, int N, int K, int act)
{
    __shared__ __align__(16) float As[64][36];
    __shared__ __align__(16) float Wp[16][64][2];

    const int tid  = threadIdx.x;
    const int lane = tid & 31;
    const int wave = tid >> 5;
    const int wr   = wave & 3;     // 4 row tiles of 16
    const int wc   = wave >> 2;    // 2 col tiles of 32
    const int bm   = blockIdx.y * 64;
    const int bn   = blockIdx.x * 64;

    const int lg   = lane >> 4;    // lane group (0/1)
    const int lm   = lane & 15;
    const int kg   = lg * 2;       // K offset of this lane group inside a k4 step

    v8f acc0 = {0.f,0.f,0.f,0.f,0.f,0.f,0.f,0.f};
    v8f acc1 = {0.f,0.f,0.f,0.f,0.f,0.f,0.f,0.f};

    for (int k0 = 0; k0 < K; k0 += 32) {
        // prefetch next K-slab (global_prefetch_b8)
        if (k0 + 32 < K) {
            __builtin_prefetch(A + (size_t)(bm + (tid & 63)) * K + (k0 + 32) + ((tid >> 6) << 3), 0, 0);
            __builtin_prefetch(W + (size_t)(k0 + 32 + (tid & 31)) * N + bn + ((tid >> 5) << 3), 0, 0);
        }
        // A tile 64x32: raw copy -> async global->LDS b128 (512 float4, 2/thread)
        #pragma unroll
        for (int i = 0; i < 2; ++i) {
            int idx = tid + i * 256;
            int r = idx >> 3, c4 = idx & 7;
            async_copy16(A + (size_t)(bm + r) * K + k0 + c4 * 4, &As[r][c4 * 4]);
        }
        // W tile 32x64: reformat into k-paired layout (regular loads)
        #pragma unroll
        for (int i = 0; i < 2; ++i) {
            int idx = tid + i * 256;
            int r = idx >> 4, c = idx & 15;
            const float4 vw = *(const float4*)(W + (size_t)(k0 + r) * N + bn + c * 4);
            const int p = r >> 1, q = r & 1;
            Wp[p][c*4+0][q] = vw.x; Wp[p][c*4+1][q] = vw.y;
            Wp[p][c*4+2][q] = vw.z; Wp[p][c*4+3][q] = vw.w;
        }
        async_join();

        const int m0 = wr * 16, n0 = wc * 32;
        const int ma = m0 + lm;
        #pragma unroll
        for (int kk = 0; kk < 32; kk += 4) {
            const int p = (kk >> 1) + lg;                   // (kk+kg)/2
            v2f a  = *(const v2f*)&As[ma][kk + kg];
            v2f b0 = *(const v2f*)&Wp[p][n0 + lm][0];
            v2f b1 = *(const v2f*)&Wp[p][n0 + 16 + lm][0];
            acc0 = __builtin_amdgcn_wmma_f32_16x16x4_f32(false, a, false, b0, (short)0, acc0, false, false);
            acc1 = __builtin_amdgcn_wmma_f32_16x16x4_f32(false, a, false, b1, (short)0, acc1, false, false);
        }
        __syncthreads();
    }

    // epilogue: bias + activation + optional residuals
    #pragma unroll
    for (int v = 0; v < 8; ++v) {
        const int row = bm + wr * 16 + v + 8 * lg;
        #pragma unroll
        for (int t = 0; t < 2; ++t) {
            const int col = bn + wc * 32 + t * 16 + lm;
            float val = (t == 0 ? acc0[v] : acc1[v]) + bias[col];
            if (act == ACT_TANH1)      val = tanhf(val) + 1.0f;
            else if (act == ACT_GELU)  val = 0.5f * val * (1.0f + erff(val * 0.70710678118654752f));
            const size_t off = (size_t)row * N + col;
            if (r1) val += r1[off];
            if (r2) val += r2[off];
            C[off] = val;
        }
    }
}

// ---------------------------------------------------------------------------
// Row LayerNorm over N=1024 (in-place safe: values cached in registers).
// ---------------------------------------------------------------------------
__global__ __launch_bounds__(256)
void layernorm_kernel(const float* __restrict__ X, const float* __restrict__ g,
                      const float* __restrict__ beta, float* __restrict__ Y)
{
    const int N = 1024;
    const size_t row = blockIdx.x;
    const int tid = threadIdx.x;
    float v[4], s = 0.f, s2 = 0.f;
    #pragma unroll
    for (int j = 0; j < 4; ++j) {
        float t = X[row * N + tid + j * 256];
        v[j] = t; s += t; s2 += t * t;
    }
    __shared__ float r1[256], r2[256];
    r1[tid] = s; r2[tid] = s2;
    __syncthreads();
    for (int o = 128; o > 0; o >>= 1) {
        if (tid < o) { r1[tid] += r1[tid + o]; r2[tid] += r2[tid + o]; }
        __syncthreads();
    }
    const float mean = r1[0] * (1.0f / N);
    const float var  = r2[0] * (1.0f / N) - mean * mean;
    const float rstd = rsqrtf(var + 1e-5f);
    #pragma unroll
    for (int j = 0; j < 4; ++j) {
        int c = tid + j * 256;
        Y[row * N + c] = (v[j] - mean) * rstd * g[c] + beta[c];
    }
}

// ---------------------------------------------------------------------------
// q_probe[b,h,d] = mean_l Q[b,l,h*64+d] / sqrt(64)
// ---------------------------------------------------------------------------
__global__ __launch_bounds__(256)
void qprobe_kernel(const float* __restrict__ Q, float* __restrict__ qp, int L, int H)
{
    const int bh = blockIdx.x, b = bh / H, h = bh % H, d = H * 64;
    const int tid = threadIdx.x, dd = tid & 63, ch = tid >> 6;
    float s = 0.f;
    for (int l = ch; l < L; l += 4)
        s += Q[((size_t)b * L + l) * d + h * 64 + dd];
    __shared__ float red[256];
    red[tid] = s;
    __syncthreads();
    if (tid < 128) red[tid] += red[tid + 128];
    __syncthreads();
    if (tid < 64) qp[(size_t)bh * 64 + tid] = (red[tid] + red[tid + 64]) * (1.0f / L) * 0.125f;
}

// ---------------------------------------------------------------------------
// Per (b,h): probe-softmax over Lk=L+1 (zero token analytic), then
//   KV[d,e] = sum_k w_k*phiK[k,d]*V[k,e] ; Ksum[d] = sum_k w_k*phiK[k,d]
// Zero token: K=V=0 -> logit 0, phi = tanh(bfk)+1, contributes only to Ksum.
// phiK / V token slabs staged raw via async global->LDS; weights applied at
// consume time.
// ---------------------------------------------------------------------------
__global__ __launch_bounds__(256)
void attn_kv_kernel(const float* __restrict__ Kc, const float* __restrict__ V,
                    const float* __restrict__ phiK, const float* __restrict__ bfk,
                    const float* __restrict__ qprobe,
                    float* __restrict__ KV, float* __restrict__ Ksum, int L, int H)
{
    const int bh = blockIdx.x, b = bh / H, h = bh % H, d = H * 64;
    const int tid = threadIdx.x;
    __shared__ float qp[64];
    __shared__ float sc[4097];      // L+1 logits -> softmax weights
    __shared__ float red[256];
    __shared__ __align__(16) float stgPK[8 * 64];
    __shared__ __align__(16) float stgV [8 * 64];

    if (tid < 64) qp[tid] = qprobe[(size_t)bh * 64 + tid];
    __syncthreads();

    // logits (sc[0] = zero-token logit 0)
    for (int i = tid; i < L + 1; i += 256) {
        float s = 0.f;
        if (i > 0) {
            const float* kr = Kc + ((size_t)b * L + (i - 1)) * d + h * 64;
            for (int j = 0; j < 64; ++j) s += qp[j] * kr[j];
        }
        sc[i] = s;
    }
    __syncthreads();

    // max
    float mx = -INFINITY;
    for (int i = tid; i < L + 1; i += 256) mx = fmaxf(mx, sc[i]);
    red[tid] = mx;
    __syncthreads();
    for (int o = 128; o > 0; o >>= 1) {
        if (tid < o) red[tid] = fmaxf(red[tid], red[tid + o]);
        __syncthreads();
    }
    mx = red[0];
    __syncthreads();

    // exp + sum
    float sm = 0.f;
    for (int i = tid; i < L + 1; i += 256) {
        float e = expf(sc[i] - mx);
        sc[i] = e; sm += e;
    }
    red[tid] = sm;
    __syncthreads();
    for (int o = 128; o > 0; o >>= 1) {
        if (tid < o) red[tid] += red[tid + o];
        __syncthreads();
    }
    const float rZ = 1.0f / red[0];

    // KV / Ksum accumulation, 8 tokens staged per step
    float acc[16];
    #pragma unroll
    for (int j = 0; j < 16; ++j) acc[j] = 0.f;
    float ks = 0.f;
    const size_t rowbase = (size_t)b * L * d + h * 64;

    for (int k0 = 0; k0 < L; k0 += 8) {
        __syncthreads();                         // protect stg reuse
        if (tid < 128) {                         // waves 0-3 only (uniform)
            int r = tid >> 4, c4 = tid & 15;
            const size_t go = rowbase + (size_t)(k0 + r) * d + c4 * 4;
            async_copy16(phiK + go, &stgPK[r * 64 + c4 * 4]);
            async_copy16(V    + go, &stgV [r * 64 + c4 * 4]);
        }
        async_join();

        float w[8];
        #pragma unroll
        for (int r = 0; r < 8; ++r) w[r] = sc[k0 + r + 1] * rZ;

        #pragma unroll
        for (int j = 0; j < 16; ++j) {
            int idx = tid + j * 256;
            int dd = idx >> 6, e = idx & 63;
            float a = 0.f;
            #pragma unroll
            for (int r = 0; r < 8; ++r)
                a = fmaf(stgPK[r * 64 + dd] * w[r], stgV[r * 64 + e], a);
            acc[j] += a;
        }
        if (tid < 64) {
            #pragma unroll
            for (int r = 0; r < 8; ++r) ks = fmaf(w[r], stgPK[r * 64 + tid], ks);
        }
    }
    #pragma unroll
    for (int j = 0; j < 16; ++j)
        KV[(size_t)bh * 4096 + tid + j * 256] = acc[j];
    if (tid < 64) {
        float pk0 = tanhf(bfk[h * 64 + tid]) + 1.0f;   // phi of zero token
        Ksum[(size_t)bh * 64 + tid] = ks + sc[0] * rZ * pk0;
    }
}

// ---------------------------------------------------------------------------
// out[b,l,h*64+e] = (phiQ[b,l,h*64,:] . KV[b,h,:,e]) / (phiQ . Ksum + 1e-6)
// ---------------------------------------------------------------------------
__global__ __launch_bounds__(256)
void numden_kernel(const float* __restrict__ phiQ, const float* __restrict__ KV,
                   const float* __restrict__ Ksum, float* __restrict__ out,
                   int L, int H)
{
    const int d = H * 64;
    const size_t row = blockIdx.x;           // b*L + l
    const int b = (int)(row / L);
    const int tid = threadIdx.x;
    __shared__ __align__(16) float pq[1024];
    __shared__ float den[16];

    async_copy16(phiQ + row * d + tid * 4, &pq[tid * 4]);
    async_join();

    if (tid < H) {
        float s = 1e-6f;
        const float* ksp = Ksum + ((size_t)b * H + tid) * 64;
        for (int j = 0; j < 64; ++j) s += pq[tid * 64 + j] * ksp[j];
        den[tid] = s;
    }
    __syncthreads();
    for (int i = tid; i < d; i += 256) {
        int h = i >> 6, e = i & 63;
        const float* kv = KV + ((size_t)b * H + h) * 4096;
        float s = 0.f;
        for (int j = 0; j < 64; ++j) s += pq[h * 64 + j] * kv[j * 64 + e];
        out[row * d + i] = s / den[h];
    }
}

// ---------------------------------------------------------------------------
extern "C" void kernel_launch(void* const* d_in, const int* in_sizes, int n_in,
                              void* d_out, int out_size, void* d_ws, size_t ws_size,
                              hipStream_t stream)
{
    (void)in_sizes; (void)n_in; (void)out_size; (void)ws_size;

    const int B = 4, L = 4096, d = 1024, mlp = 4096, H = 16;
    const int M = B * L;                                  // 16384 rows

    const float* x     = (const float*)d_in[0];
    const float* Wq    = (const float*)d_in[1];
    const float* bq    = (const float*)d_in[2];
    const float* gq    = (const float*)d_in[3];
    const float* betaq = (const float*)d_in[4];
    const float* Wk    = (const float*)d_in[5];
    const float* bk    = (const float*)d_in[6];
    const float* gk    = (const float*)d_in[7];
    const float* betak = (const float*)d_in[8];
    const float* Wv    = (const float*)d_in[9];
    const float* bv    = (const float*)d_in[10];
    const float* gv    = (const float*)d_in[11];
    const float* betav = (const float*)d_in[12];
    const float* Wfq   = (const float*)d_in[13];
    const float* bfq   = (const float*)d_in[14];
    const float* Wfk   = (const float*)d_in[15];
    const float* bfk   = (const float*)d_in[16];
    const float* ga    = (const float*)d_in[17];
    const float* ba    = (const float*)d_in[18];
    const float* W1    = (const float*)d_in[19];
    const float* b1    = (const float*)d_in[20];
    const float* W2    = (const float*)d_in[21];
    const float* b2    = (const float*)d_in[22];
    float* out = (float*)d_out;

    // workspace layout (floats). Q..phiK are contiguous 256MB reused as the
    // MLP hidden (h1) once dead. Total ~386 MB.
    float* wsf = (float*)d_ws;
    const size_t szMd = (size_t)M * d;
    float* Q    = wsf;
    float* Kb   = Q    + szMd;
    float* Vb   = Kb   + szMd;
    float* pK   = Vb   + szMd;
    float* pQ   = pK   + szMd;
    float* attn = pQ   + szMd;
    float* qpb  = attn + szMd;                 // B*H*64
    float* KVb  = qpb  + (size_t)B * H * 64;   // B*H*64*64
    float* KsB  = KVb  + (size_t)B * H * 4096; // B*H*64
    float* h1   = Q;                           // alias: M*mlp == 4*szMd

    const dim3 blk(256);
    const dim3 gD(d / 64, M / 64);             // [M,1024] outputs
    const dim3 gMLP(mlp / 64, M / 64);         // [M,4096] outputs

    // Q/K/V projections
    gemm_f32_wmma_kernel<<<gD, blk, 0, stream>>>(x, Wq, bq, nullptr, nullptr, Q,  M, d, d, ACT_NONE);
    gemm_f32_wmma_kernel<<<gD, blk, 0, stream>>>(x, Wk, bk, nullptr, nullptr, Kb, M, d, d, ACT_NONE);
    gemm_f32_wmma_kernel<<<gD, blk, 0, stream>>>(x, Wv, bv, nullptr, nullptr, Vb, M, d, d, ACT_NONE);
    // LayerNorms (in place)
    layernorm_kernel<<<M, blk, 0, stream>>>(Q,  gq, betaq, Q);
    layernorm_kernel<<<M, blk, 0, stream>>>(Kb, gk, betak, Kb);
    layernorm_kernel<<<M, blk, 0, stream>>>(Vb, gv, betav, Vb);
    // feature maps phi = tanh(.)+1
    gemm_f32_wmma_kernel<<<gD, blk, 0, stream>>>(Q,  Wfq, bfq, nullptr, nullptr, pQ, M, d, d, ACT_TANH1);
    gemm_f32_wmma_kernel<<<gD, blk, 0, stream>>>(Kb, Wfk, bfk, nullptr, nullptr, pK, M, d, d, ACT_TANH1);
    // probe + gated KV / Ksum
    qprobe_kernel<<<B * H, blk, 0, stream>>>(Q, qpb, L, H);
    attn_kv_kernel<<<B * H, blk, 0, stream>>>(Kb, Vb, pK, bfk, qpb, KVb, KsB, L, H);
    // linear-attention output
    numden_kernel<<<M, blk, 0, stream>>>(pQ, KVb, KsB, attn, L, H);
    // attn norm (in place)
    layernorm_kernel<<<M, blk, 0, stream>>>(attn, ga, ba, attn);
    // MLP with fused residuals: out = gelu(a@W1+b1)@W2 + b2 + a + x
    gemm_f32_wmma_kernel<<<gMLP, blk, 0, stream>>>(attn, W1, b1, nullptr, nullptr, h1, M, mlp, d, ACT_GELU);
    gemm_f32_wmma_kernel<<<gD, blk, 0, stream>>>(h1, W2, b2, attn, x, out, M, d, mlp, ACT_NONE);
}